// ScaledDotProductWithEdgeAttention_15487652069889
// MI455X (gfx1250) — compile-verified
//
#include <hip/hip_runtime.h>

// B=16, H=8, N=64, L=64, DK=64 ; TEMPERATURE=8 ; MASK_FILL=-2^15
#define L_DIM 64
#define DK_DIM 64
#define TEMP_INV 0.125f
#define MASK_FILL_F (-32768.0f)

typedef float v2f __attribute__((ext_vector_type(2)));
typedef float v8f __attribute__((ext_vector_type(8)));

__global__ __launch_bounds__(128) void sdpa_edge_attn_kernel(
    const float* __restrict__ q, const float* __restrict__ k,
    const float* __restrict__ v, const int* __restrict__ mask,
    float* __restrict__ out, int HN) {
  const int p = blockIdx.x;        // (b,h,n) flat index, 8192 total
  const int b = p / HN;            // batch index for mask
  const int tid  = threadIdx.x;
  const int w    = tid >> 5;       // wave id 0..3 (wave32)
  const int lane = tid & 31;
  const int half = lane >> 4;      // 0: K=0,1  1: K=2,3  (f32 16x16x4 A/B layout)
  const int l16  = lane & 15;
  const int koff = half * 2;

  __shared__ float ldsQ[L_DIM * DK_DIM]; // later reused as P (per-wave slab)
  __shared__ float ldsK[L_DIM * DK_DIM];
  __shared__ float ldsV[L_DIM * DK_DIM];

  const size_t base = (size_t)p * (L_DIM * DK_DIM);

  // Cooperative staged load: 4096 floats per matrix, 128 thr x 8 iters x float4
#pragma unroll
  for (int i = 0; i < 8; ++i) {
    const int off = i * 512 + tid * 4;
    *(float4*)&ldsQ[off] = *(const float4*)&q[base + off];
    *(float4*)&ldsK[off] = *(const float4*)&k[base + off];
    *(float4*)&ldsV[off] = *(const float4*)&v[base + off];
  }
  __syncthreads();

  // ---------------- GEMM1: S = Q * K^T  (wave w -> rows 16w..16w+15) --------
  const int arow = w * 16 + l16;
  v8f acc[4];
#pragma unroll
  for (int t = 0; t < 4; ++t) acc[t] = (v8f)0.0f;

#pragma unroll
  for (int d0 = 0; d0 < DK_DIM; d0 += 4) {
    // A fragment: Q[arow][d0+koff .. +1]   (contiguous -> ds_load_b64)
    const v2f a = *(const v2f*)&ldsQ[arow * DK_DIM + d0 + koff];
#pragma unroll
    for (int t = 0; t < 4; ++t) {
      // B fragment: B[k][n] = K[t*16+n][d0+k]  -> K[t*16+l16][d0+koff .. +1]
      const v2f bv = *(const v2f*)&ldsK[(t * 16 + l16) * DK_DIM + d0 + koff];
      acc[t] = __builtin_amdgcn_wmma_f32_16x16x4_f32(
          false, a, false, bv, (short)0, acc[t], false, false);
    }
  }

  // ---------------- mask + scale + row softmax (in C layout) ----------------
  // C layout: VGPR slot i holds row (i + half*8), column t*16 + l16
  const int* __restrict__ mrow = mask + (size_t)b * (L_DIM * L_DIM);
#pragma unroll
  for (int i = 0; i < 8; ++i) {
    const int qrow = w * 16 + half * 8 + i;
#pragma unroll
    for (int t = 0; t < 4; ++t) {
      float s = acc[t][i] * TEMP_INV;
      const int m = mrow[qrow * L_DIM + t * 16 + l16];
      acc[t][i] = (m == 0) ? MASK_FILL_F : s;
    }
    // row max: 4 tiles in-register, then across the 16 lanes of each half
    float mx = fmaxf(fmaxf(acc[0][i], acc[1][i]), fmaxf(acc[2][i], acc[3][i]));
    mx = fmaxf(mx, __shfl_xor(mx, 1, 32));
    mx = fmaxf(mx, __shfl_xor(mx, 2, 32));
    mx = fmaxf(mx, __shfl_xor(mx, 4, 32));
    mx = fmaxf(mx, __shfl_xor(mx, 8, 32));
    float sum = 0.0f;
#pragma unroll
    for (int t = 0; t < 4; ++t) {
      const float e = __expf(acc[t][i] - mx);
      acc[t][i] = e;
      sum += e;
    }
    sum += __shfl_xor(sum, 1, 32);
    sum += __shfl_xor(sum, 2, 32);
    sum += __shfl_xor(sum, 4, 32);
    sum += __shfl_xor(sum, 8, 32);
    const float inv = 1.0f / sum;
    // store normalized P over this wave's (now dead) Q slab
#pragma unroll
    for (int t = 0; t < 4; ++t)
      ldsQ[qrow * DK_DIM + t * 16 + l16] = acc[t][i] * inv;
  }
  __syncthreads();

  // ---------------- GEMM2: O = P * V ----------------------------------------
  v8f oacc[4];
#pragma unroll
  for (int t = 0; t < 4; ++t) oacc[t] = (v8f)0.0f;

#pragma unroll
  for (int k0 = 0; k0 < L_DIM; k0 += 4) {
    // A fragment: P[arow][k0+koff .. +1]
    const v2f a = *(const v2f*)&ldsQ[arow * DK_DIM + k0 + koff];
#pragma unroll
    for (int t = 0; t < 4; ++t) {
      // B fragment: B[k][n] = V[k0+k][t*16+n]  (stride DK between the 2 K's)
      v2f bv;
      bv.x = ldsV[(k0 + koff + 0) * DK_DIM + t * 16 + l16];
      bv.y = ldsV[(k0 + koff + 1) * DK_DIM + t * 16 + l16];
      oacc[t] = __builtin_amdgcn_wmma_f32_16x16x4_f32(
          false, a, false, bv, (short)0, oacc[t], false, false);
    }
  }

  // ---------------- store O (C layout, lane-coalesced) ----------------------
  float* __restrict__ outp = out + base;
#pragma unroll
  for (int i = 0; i < 8; ++i) {
    const int qrow = w * 16 + half * 8 + i;
#pragma unroll
    for (int t = 0; t < 4; ++t)
      outp[qrow * DK_DIM + t * 16 + l16] = oacc[t][i];
  }
}

extern "C" void kernel_launch(void* const* d_in, const int* in_sizes, int n_in,
                              void* d_out, int out_size, void* d_ws, size_t ws_size,
                              hipStream_t stream) {
  const float* q   = (const float*)d_in[0];
  const float* k   = (const float*)d_in[1];
  const float* v   = (const float*)d_in[2];
  const int* mask  = (const int*)d_in[3];
  float* out       = (float*)d_out;

  // total heads = B*H*N = in_sizes[0] / (L*DK); H*N = 512 for this problem
  const int total = in_sizes[0] / (L_DIM * DK_DIM); // 8192
  const int HN = 512;                               // H*N (mask broadcast span)

  sdpa_edge_attn_kernel<<<dim3(total), dim3(128), 0, stream>>>(
      q, k, v, mask, out, HN);
}